// RNN_88012469829723
// MI455X (gfx1250) — compile-verified
//
#include <hip/hip_runtime.h>

typedef __attribute__((ext_vector_type(2))) float v2f;
typedef __attribute__((ext_vector_type(8))) float v8f;

#define NN      1024      // N_NEURONS (= K = N of the GEMM)
#define BB      2048      // BATCH (= M of the GEMM)
#define NSTEPS  32
#define EPS     0.01f

#define KTILE   16        // K staged per LDS chunk (4 WMMA sub-chunks)
#define LDSW    18        // padded LDS row stride (floats): even (8B align for b64),
                          // 18 mod 64 spreads banks across the 16-row fragment read

// ---------------------------------------------------------------------------
// Zero-fill h_0 (first state slab of d_out). float4 stores.
// ---------------------------------------------------------------------------
__global__ __launch_bounds__(256) void rnn_zero(float4* __restrict__ p, int n4) {
    int i = blockIdx.x * blockDim.x + threadIdx.x;
    if (i < n4) p[i] = make_float4(0.f, 0.f, 0.f, 0.f);
}

// ---------------------------------------------------------------------------
// One recurrence step: h_out = h_in + EPS * tanh(h_in @ W + b)
// GEMM M=2048, N=1024, K=1024 in fp32 via V_WMMA_F32_16X16X4_F32.
//
// Workgroup: 256 threads = 8 waves, 2 (rows) x 4 (cols) over a 128x128 output
// tile; each wave owns a 64x32 tile = 4x2 WMMA accumulators.
//
// A path: cooperative coalesced global load of the 128x16 A tile into padded
//         LDS (each thread: 8 contiguous floats = half a row-chunk), then
//         per-wave ds_load_b64 fragment reads in WMMA layout.
// B path: direct global loads (consecutive columns per lane = full lines; W
//         is 4 MB and stays resident in L2).
//
// Fragment layouts (ISA 7.12.2, 32-bit data, wave32):
//   A (16x4 MxK):  lanes 0-15: M=lane,    K pair {0,1}; lanes 16-31: K pair {2,3}
//   B (4x16 KxN):  lanes 0-15: N=lane,    K pair {0,1}; lanes 16-31: K pair {2,3}
//   C/D (16x16):   VGPR v: lanes 0-15 -> M=v, lanes 16-31 -> M=v+8, N=lane%16
// ---------------------------------------------------------------------------
__global__ __launch_bounds__(256) void rnn_step(const float* __restrict__ h_in,
                                                const float* __restrict__ W,
                                                const float* __restrict__ bias,
                                                float*       __restrict__ h_out) {
    __shared__ float As[128 * LDSW];             // 9216 B

    const int tid   = threadIdx.x;
    const int lane  = tid & 31;
    const int wave  = tid >> 5;
    const int wrow  = wave >> 2;                 // 0..1
    const int wcol  = wave & 3;                  // 0..3
    const int rowB  = blockIdx.y * 128;                 // WG M origin
    const int col0  = blockIdx.x * 128 + wcol * 32;     // wave N origin
    const int wR0   = wrow * 64;                        // wave M origin in tile

    const bool lolane = (lane < 16);
    const int  l15    = lane & 15;

    // Cooperative A-tile load mapping: 2 threads per row, 8 floats each (32B)
    const int    ldRow = tid >> 1;               // 0..127
    const int    ldK   = (tid & 1) * 8;          // 0 or 8
    const float* gA    = h_in + (size_t)(rowB + ldRow) * NN + ldK;
    float*       sA    = As + ldRow * LDSW + ldK;

    v8f acc[4][2];
#pragma unroll
    for (int i = 0; i < 4; ++i)
#pragma unroll
        for (int j = 0; j < 2; ++j)
            acc[i][j] = (v8f)0.0f;

    // Prefetch first A chunk (coalesced 2x float4 per thread)
    float4 pre0 = *(const float4*)(gA + 0);
    float4 pre1 = *(const float4*)(gA + 4);

    for (int k0 = 0; k0 < NN; k0 += KTILE) {
        __syncthreads();                         // LDS consumers of prev chunk done
        // reg -> LDS (float2 stores: padded stride only guarantees 8B align)
        ((float2*)sA)[0] = make_float2(pre0.x, pre0.y);
        ((float2*)sA)[1] = make_float2(pre0.z, pre0.w);
        ((float2*)sA)[2] = make_float2(pre1.x, pre1.y);
        ((float2*)sA)[3] = make_float2(pre1.z, pre1.w);
        __syncthreads();                         // tile visible to all waves

        // Prefetch next chunk while computing this one
        if (k0 + KTILE < NN) {
            pre0 = *(const float4*)(gA + k0 + KTILE);
            pre1 = *(const float4*)(gA + k0 + KTILE + 4);
        }

        // 4 WMMA sub-chunks of K=4  ->  32 matrix ops per wave per chunk
#pragma unroll
        for (int kc = 0; kc < KTILE; kc += 4) {
            const int kk = kc + (lolane ? 0 : 2);        // lane-half K pair

            // A fragments from LDS (ds_load_b64 per fragment)
            v2f a[4];
#pragma unroll
            for (int i = 0; i < 4; ++i) {
                const float2 t =
                    *(const float2*)(As + (wR0 + 16 * i + l15) * LDSW + kk);
                a[i].x = t.x;
                a[i].y = t.y;
            }

            // B fragments direct from global (L2-resident W)
            const int kg = k0 + kk;
            v2f bf[2];
#pragma unroll
            for (int j = 0; j < 2; ++j) {
                const int n = col0 + 16 * j + l15;
                bf[j].x = W[(size_t)(kg + 0) * NN + n];
                bf[j].y = W[(size_t)(kg + 1) * NN + n];
            }

#pragma unroll
            for (int i = 0; i < 4; ++i)
#pragma unroll
                for (int j = 0; j < 2; ++j)
                    acc[i][j] = __builtin_amdgcn_wmma_f32_16x16x4_f32(
                        /*neg_a=*/false, a[i],
                        /*neg_b=*/false, bf[j],
                        /*c_mod=*/(short)0, acc[i][j],
                        /*reuse_a=*/false, /*reuse_b=*/false);
        }
    }

    // Epilogue: h_out = h_in + EPS * tanh(acc + bias)
#pragma unroll
    for (int j = 0; j < 2; ++j) {
        const int   col = col0 + 16 * j + l15;
        const float bv  = bias[col];
#pragma unroll
        for (int i = 0; i < 4; ++i) {
#pragma unroll
            for (int v = 0; v < 8; ++v) {
                const int   row = rowB + wR0 + 16 * i + (lolane ? v : v + 8);
                const float z   = acc[i][j][v] + bv;
                const float t   = tanhf(z);
                const size_t idx = (size_t)row * NN + col;
                h_out[idx] = h_in[idx] + EPS * t;
            }
        }
    }
}

// ---------------------------------------------------------------------------
// d_in[0] = X [2048x1024] (only sets batch size; unused numerically)
// d_in[1] = W [1024x1024] fp32
// d_in[2] = b [1x1024]    fp32
// d_out   = states [33, 2048, 1024] fp32; slab t is state h_t (h_0 = 0)
// ---------------------------------------------------------------------------
extern "C" void kernel_launch(void* const* d_in, const int* in_sizes, int n_in,
                              void* d_out, int out_size, void* d_ws, size_t ws_size,
                              hipStream_t stream) {
    (void)in_sizes; (void)n_in; (void)out_size; (void)d_ws; (void)ws_size;

    const float* W    = (const float*)d_in[1];
    const float* bias = (const float*)d_in[2];
    float*       out  = (float*)d_out;

    const size_t S = (size_t)BB * NN;            // elements per state slab

    // h_0 = 0
    {
        const int n4 = (int)(S / 4);
        rnn_zero<<<(n4 + 255) / 256, 256, 0, stream>>>((float4*)out, n4);
    }

    // 32 dependent steps; each reads slab t, writes slab t+1
    dim3 grid(NN / 128, BB / 128);               // 8 x 16 workgroups
    for (int t = 0; t < NSTEPS; ++t) {
        rnn_step<<<grid, dim3(256), 0, stream>>>(out + (size_t)t * S, W, bias,
                                                 out + (size_t)(t + 1) * S);
    }
}